// EncoderGNN_89103391523116
// MI455X (gfx1250) — compile-verified
//
#include <hip/hip_runtime.h>

#define NN 50000
#define NE 800000
#define D  128
#define LN_EPS 1e-5f

typedef float v2f __attribute__((ext_vector_type(2)));
typedef float v8f __attribute__((ext_vector_type(8)));

// ---------------- zero fill ----------------
__global__ void k_fill0(float* __restrict__ p, int n) {
    int i = blockIdx.x * blockDim.x + threadIdx.x;
    if (i < n) p[i] = 0.0f;
}

// ---------------- degree count ----------------
__global__ void k_count(const long long* __restrict__ dst, float* __restrict__ cnt) {
    int e = blockIdx.x * blockDim.x + threadIdx.x;
    if (e < NE) atomicAdd(cnt + (int)dst[e], 1.0f);
}

// ---------------- edge scatter: agg[dst] += feat[src] ----------------
// one wave32 per edge; each lane moves 4 contiguous floats (b128 load + 4 f32 atomics)
__global__ void k_scatter(const long long* __restrict__ src,
                          const long long* __restrict__ dst,
                          const float* __restrict__ feat,
                          float* __restrict__ agg) {
    int wave = (blockIdx.x * blockDim.x + threadIdx.x) >> 5;
    int lane = threadIdx.x & 31;
    if (wave >= NE) return;
    int s = (int)src[wave];
    int d = (int)dst[wave];
    float4 v = *(const float4*)(feat + (size_t)s * D + lane * 4);
    float* o = agg + (size_t)d * D + lane * 4;
    atomicAdd(o + 0, v.x);
    atomicAdd(o + 1, v.y);
    atomicAdd(o + 2, v.z);
    atomicAdd(o + 3, v.w);
}

// ---------------- fused SAGE layer: out = relu((agg/cnt)@Wl^T + bl + xin@Wr^T) ----------------
// Wave computes a 16-row x 128-col strip with V_WMMA_F32_16X16X4_F32.
// A (16x4 f32, 2 VGPR): lane -> M = lane&15, K pair = 2*(lane>>4).
// B (4x16 f32, 2 VGPR): lane -> N = lane&15, K pair = 2*(lane>>4).  B[k][n] = W[n][k].
// C/D (16x16 f32, 8 VGPR): vgpr r -> M = r + 8*(lane>>4), N = lane&15.
__global__ void k_sage_gemm(const float* __restrict__ agg, const float* __restrict__ cnt,
                            const float* __restrict__ xin,
                            const float* __restrict__ Wl, const float* __restrict__ bl,
                            const float* __restrict__ Wr, float* __restrict__ out) {
    extern __shared__ float lds[];          // [2][D*D] = 128 KB of the 320 KB WGP LDS
    float* ldsWl = lds;
    float* ldsWr = lds + D * D;

    // cooperative stage of both 64 KB weight matrices into LDS (float4 = b128)
    for (int i = threadIdx.x; i < D * D / 4; i += blockDim.x) {
        ((float4*)ldsWl)[i] = ((const float4*)Wl)[i];
        ((float4*)ldsWr)[i] = ((const float4*)Wr)[i];
    }
    __syncthreads();

    const int lane = threadIdx.x & 31;
    const int wave = threadIdx.x >> 5;
    const int i0   = (blockIdx.x * 8 + wave) * 16;   // NN = 3125*16 -> no tail
    if (i0 >= NN) return;                            // wave-uniform

    const int nIdx   = lane & 15;
    const int koff   = (lane >> 4) * 2;
    const int rowoff = (lane >> 4) * 8;

    const int rA = i0 + nIdx;                        // this lane's fixed A row
    const float invc = 1.0f / fmaxf(cnt[rA], 1.0f);  // mean divisor, loop-invariant
    const float* pa = agg + (size_t)rA * D;
    const float* px = xin + (size_t)rA * D;

    v8f acc[8] = {};

    for (int kk = 0; kk < D; kk += 4) {
        const int kb = kk + koff;
        v2f am = *(const v2f*)(pa + kb);             // global b64
        am.x *= invc;
        am.y *= invc;
        v2f ax = *(const v2f*)(px + kb);
#pragma unroll
        for (int jt = 0; jt < 8; ++jt) {
            const int col = jt * 16 + nIdx;
            v2f bL = *(const v2f*)(ldsWl + col * D + kb);   // ds b64
            v2f bR = *(const v2f*)(ldsWr + col * D + kb);
            acc[jt] = __builtin_amdgcn_wmma_f32_16x16x4_f32(
                false, am, false, bL, (short)0, acc[jt], false, false);
            acc[jt] = __builtin_amdgcn_wmma_f32_16x16x4_f32(
                false, ax, false, bR, (short)0, acc[jt], false, false);
        }
    }

    // bias + relu + store (16 consecutive floats per (jt,r) per half-wave)
#pragma unroll
    for (int jt = 0; jt < 8; ++jt) {
        const int col  = jt * 16 + nIdx;
        const float bs = bl[col];
#pragma unroll
        for (int r = 0; r < 8; ++r) {
            const int row = i0 + rowoff + r;
            out[(size_t)row * D + col] = fmaxf(acc[jt][r] + bs, 0.0f);
        }
    }
}

// ---------------- LayerNorm, wave per row, in place ----------------
__global__ void k_layernorm(float* __restrict__ h,
                            const float* __restrict__ w,
                            const float* __restrict__ b) {
    int wave = (blockIdx.x * blockDim.x + threadIdx.x) >> 5;
    int lane = threadIdx.x & 31;
    if (wave >= NN) return;
    float* row = h + (size_t)wave * D;
    float4 v = *(const float4*)(row + lane * 4);

    float s = v.x + v.y + v.z + v.w;
#pragma unroll
    for (int m = 16; m; m >>= 1) s += __shfl_xor(s, m, 32);
    const float mu = s * (1.0f / D);

    float dx = v.x - mu, dy = v.y - mu, dz = v.z - mu, dw = v.w - mu;
    float q = dx * dx + dy * dy + dz * dz + dw * dw;
#pragma unroll
    for (int m = 16; m; m >>= 1) q += __shfl_xor(q, m, 32);
    const float rstd = rsqrtf(q * (1.0f / D) + LN_EPS);

    float4 wv = *(const float4*)(w + lane * 4);
    float4 bv = *(const float4*)(b + lane * 4);
    float4 o;
    o.x = dx * rstd * wv.x + bv.x;
    o.y = dy * rstd * wv.y + bv.y;
    o.z = dz * rstd * wv.z + bv.z;
    o.w = dw * rstd * wv.w + bv.w;
    *(float4*)(row + lane * 4) = o;
}

extern "C" void kernel_launch(void* const* d_in, const int* in_sizes, int n_in,
                              void* d_out, int out_size, void* d_ws, size_t ws_size,
                              hipStream_t stream) {
    const float* x        = (const float*)d_in[0];
    const long long* ei   = (const long long*)d_in[1];   // int64 edge_index [2, NE]
    const long long* src  = ei;
    const long long* dst  = ei + NE;
    const float* W1l = (const float*)d_in[2];
    const float* b1l = (const float*)d_in[3];
    const float* W1r = (const float*)d_in[4];
    const float* W2l = (const float*)d_in[5];
    const float* b2l = (const float*)d_in[6];
    const float* W2r = (const float*)d_in[7];
    const float* lnw = (const float*)d_in[8];
    const float* lnb = (const float*)d_in[9];
    float* out = (float*)d_out;

    float* ws  = (float*)d_ws;
    float* cnt = ws;                          // NN floats
    float* agg = ws + NN;                     // NN*D floats (contiguous after cnt)
    float* h1  = agg + (size_t)NN * D;        // NN*D floats

    const int zn = NN + NN * D;               // zero cnt+agg in one pass
    const dim3 blk(256);
    const size_t ldsBytes = 2u * D * D * sizeof(float);  // 128 KB
    const dim3 gemmGrid((NN + 127) / 128);               // 8 waves * 16 rows per block

    // ---- layer 1 ----
    k_fill0<<<(zn + 255) / 256, blk, 0, stream>>>(cnt, zn);
    k_count<<<(NE + 255) / 256, blk, 0, stream>>>(dst, cnt);
    k_scatter<<<NE / 8, blk, 0, stream>>>(src, dst, x, agg);
    k_sage_gemm<<<gemmGrid, blk, ldsBytes, stream>>>(agg, cnt, x, W1l, b1l, W1r, h1);

    // ---- layer 2 ----
    k_fill0<<<(NN * D + 255) / 256, blk, 0, stream>>>(agg, NN * D);
    k_scatter<<<NE / 8, blk, 0, stream>>>(src, dst, h1, agg);
    k_sage_gemm<<<gemmGrid, blk, ldsBytes, stream>>>(agg, cnt, h1, W2l, b2l, W2r, out);

    // ---- layer norm (in place on d_out) ----
    k_layernorm<<<(NN * 32 + 255) / 256, blk, 0, stream>>>(out, lnw, lnb);
}